// InvolutionNative_86431921864914
// MI455X (gfx1250) — compile-verified
//
#include <hip/hip_runtime.h>
#include <hip/hip_bf16.h>

// ---------------------------------------------------------------------------
// InvolutionNative on gfx1250 (MI455X):
//   k1: T  = ReLU(BN(W1 @ X))      -- f32 WMMA 16x16x4, fused epilogue
//   k2: Wg = W2 @ T + b2           -- f32 WMMA 16x16x4, B staged in LDS,
//                                     8 M-tile waves share one N-tile
//   k3: out[c,n] = sum_kk Wg[g*49+kk, n] * x_patch   -- VALU gather/FMA
// ---------------------------------------------------------------------------

typedef __attribute__((ext_vector_type(2))) float v2f;
typedef __attribute__((ext_vector_type(8))) float v8f;

#define BATCH 4
#define CIN   256
#define CMID  64          // C/4
#define HW    4096        // 64*64
#define WIDTH 64
#define NPIX  (BATCH * HW)   // 16384
#define KTAP  7
#define KK    49
#define NGRP  16
#define MOUT  (KK * NGRP)    // 784
#define MT2   49             // M-tiles in conv2
#define EPS   1e-5f

// ---------------------------------------------------------------------------
// Kernel 1: T[64, NPIX] = relu(bn(W1[64,256] @ X[256, NPIX] + b1))
// One wave per 16x16 output tile. Mtiles=4, Ntiles=1024.
// ---------------------------------------------------------------------------
__global__ __launch_bounds__(256) void k1_conv1_bn_relu(
    const float* __restrict__ x,   const float* __restrict__ w1,
    const float* __restrict__ b1,  const float* __restrict__ gamma,
    const float* __restrict__ beta,const float* __restrict__ mean,
    const float* __restrict__ var, float* __restrict__ T)
{
    const int lane = threadIdx.x & 31;
    const int wave = threadIdx.x >> 5;
    const int tile = blockIdx.x * 8 + wave;          // 4096 tiles
    const int mt   = tile & 3;                       // 4 M-tiles
    const int nt   = tile >> 2;                      // 1024 N-tiles
    const int m_base = mt * 16;
    const int n_base = nt * 16;
    const int half = lane >> 4;                      // 0 | 1
    const int l16  = lane & 15;

    // column n = n_base + l16 ; n -> (b, p); 16-tile never crosses batch
    const int b_idx = n_base / HW;
    const int p     = (n_base % HW) + l16;
    const float* xcol = x + (size_t)b_idx * CIN * HW + p;             // + c*HW
    const float* arow = w1 + (size_t)(m_base + l16) * CIN + 2 * half; // + k

    v8f acc = {};
#pragma unroll 4
    for (int k = 0; k < CIN; k += 4) {
        v2f a, b;
        a.x = arow[k];                                  // A[m][k+2h+0]
        a.y = arow[k + 1];                              // A[m][k+2h+1]
        b.x = xcol[(size_t)(k + 2 * half)     * HW];    // B[k+2h+0][n]
        b.y = xcol[(size_t)(k + 2 * half + 1) * HW];    // B[k+2h+1][n]
        acc = __builtin_amdgcn_wmma_f32_16x16x4_f32(
                  false, a, false, b, (short)0, acc, false, false);
    }

    // epilogue: bias + BN(eval) + ReLU, then store D tile
    const int ncol = n_base + l16;
#pragma unroll
    for (int r = 0; r < 8; ++r) {
        const int m = m_base + r + 8 * half;
        float t = acc[r] + b1[m];
        t = gamma[m] * (t - mean[m]) * rsqrtf(var[m] + EPS) + beta[m];
        t = fmaxf(t, 0.0f);
        T[(size_t)m * NPIX + ncol] = t;
    }
}

// ---------------------------------------------------------------------------
// Kernel 2: Wg[784, NPIX] = W2[784,64] @ T[64, NPIX] + b2
// Block = 8 waves sharing one N-tile; T tile (64x16 = 4KB) staged in LDS once
// and fed to all waves' WMMA B operands (cuts T read amplification 49x -> 7x).
// grid = 1024 N-tiles * 7 M-chunks.
// ---------------------------------------------------------------------------
__global__ __launch_bounds__(256) void k2_conv2(
    const float* __restrict__ T, const float* __restrict__ w2,
    const float* __restrict__ b2, float* __restrict__ wg)
{
    __shared__ float bt[CMID * 16];                  // [k][n], n fastest

    const int nt    = blockIdx.x & 1023;             // 1024 N-tiles
    const int chunk = blockIdx.x >> 10;              // 7 chunks of 8 M-tiles
    const int n_base = nt * 16;

    // cooperative coalesced fill of the shared B tile
#pragma unroll
    for (int i = threadIdx.x; i < CMID * 16; i += 256) {
        const int k = i >> 4;
        const int n = i & 15;
        bt[i] = T[(size_t)k * NPIX + n_base + n];
    }
    __syncthreads();

    const int lane = threadIdx.x & 31;
    const int wave = threadIdx.x >> 5;
    const int mt   = chunk * 8 + wave;               // wave-uniform guard
    if (mt >= MT2) return;                           // (EXEC stays all-ones)

    const int m_base = mt * 16;
    const int half = lane >> 4;
    const int l16  = lane & 15;

    const float* arow = w2 + (size_t)(m_base + l16) * CMID + 2 * half; // + k
    const float* bcol = bt + l16;                                      // + k*16

    v8f acc = {};
#pragma unroll 4
    for (int k = 0; k < CMID; k += 4) {
        v2f a, b;
        a.x = arow[k];
        a.y = arow[k + 1];
        b.x = bcol[(k + 2 * half)     * 16];         // ds_load
        b.y = bcol[(k + 2 * half + 1) * 16];         // ds_load
        acc = __builtin_amdgcn_wmma_f32_16x16x4_f32(
                  false, a, false, b, (short)0, acc, false, false);
    }

    const int ncol = n_base + l16;
#pragma unroll
    for (int r = 0; r < 8; ++r) {
        const int m = m_base + r + 8 * half;
        wg[(size_t)m * NPIX + ncol] = acc[r] + b2[m];
    }
}

// ---------------------------------------------------------------------------
// Kernel 3: involution gather.
// out[b, c, y, x] = sum_{kh,kw} Wg[g*49 + kh*7+kw, b*HW + y*64+x]
//                              * x[b, c, y+kh-3, x+kw-3]   (zero pad)
// One thread per output element; adjacent threads = adjacent pixels
// (coalesced Wg column reads; x neighborhoods hit L2/WGP$).
// ---------------------------------------------------------------------------
__global__ __launch_bounds__(256) void k3_involution(
    const float* __restrict__ x, const float* __restrict__ wg,
    float* __restrict__ out)
{
    const int idx = blockIdx.x * 256 + threadIdx.x;   // (b, c, p), p fastest
    const int p  = idx & (HW - 1);
    const int c  = (idx >> 12) & (CIN - 1);
    const int b  = idx >> 20;
    const int yy = p >> 6;
    const int xx = p & (WIDTH - 1);
    const int g  = c >> 4;

    const float* wcol = wg + (size_t)g * KK * NPIX + (size_t)b * HW + p; // + kk*NPIX
    const float* xc   = x + ((size_t)b * CIN + c) * HW;

    float acc = 0.0f;
#pragma unroll
    for (int kh = 0; kh < KTAP; ++kh) {
        const int y2 = yy + kh - 3;
        const bool yok = (y2 >= 0) && (y2 < WIDTH);
#pragma unroll
        for (int kw = 0; kw < KTAP; ++kw) {
            const int x2 = xx + kw - 3;
            const bool ok = yok && (x2 >= 0) && (x2 < WIDTH);
            const float xv = ok ? xc[y2 * WIDTH + x2] : 0.0f;
            acc = fmaf(wcol[(size_t)(kh * KTAP + kw) * NPIX], xv, acc);
        }
    }
    out[idx] = acc;
}

// ---------------------------------------------------------------------------
extern "C" void kernel_launch(void* const* d_in, const int* in_sizes, int n_in,
                              void* d_out, int out_size, void* d_ws, size_t ws_size,
                              hipStream_t stream)
{
    (void)in_sizes; (void)n_in; (void)out_size; (void)ws_size;

    const float* x     = (const float*)d_in[0];
    const float* w1    = (const float*)d_in[1];
    const float* b1    = (const float*)d_in[2];
    const float* gamma = (const float*)d_in[3];
    const float* beta  = (const float*)d_in[4];
    const float* mean  = (const float*)d_in[5];
    const float* var   = (const float*)d_in[6];
    const float* w2    = (const float*)d_in[7];
    const float* b2    = (const float*)d_in[8];
    float* out = (float*)d_out;

    // workspace: T [64, 16384] then Wg [784, 16384], both f32 (~56 MB total)
    float* T  = (float*)d_ws;
    float* Wg = T + (size_t)CMID * NPIX;

    // k1: 4 Mtiles * 1024 Ntiles = 4096 waves -> 512 blocks of 8 waves
    k1_conv1_bn_relu<<<512, 256, 0, stream>>>(x, w1, b1, gamma, beta, mean, var, T);

    // k2: 1024 N-tiles * 7 chunks of 8 M-tiles -> 7168 blocks
    k2_conv2<<<7168, 256, 0, stream>>>(T, w2, b2, Wg);

    // k3: 4*256*4096 = 4194304 threads -> 16384 blocks
    k3_involution<<<16384, 256, 0, stream>>>(x, Wg, out);
}